// BeamSearchSummarizationModel_34162169873093
// MI455X (gfx1250) — compile-verified
//
#include <hip/hip_runtime.h>
#include <math.h>

typedef __attribute__((ext_vector_type(2))) float v2f;
typedef __attribute__((ext_vector_type(8))) float v8f;

#define SLEN 512
#define BATCH 8
#define EMB 256
#define HID 512
#define G4H 2048
#define VOC 32000
#define NBEAM 4
#define NSTEP 12
#define TLEN 13

__device__ __forceinline__ float sigm(float x) { return 1.0f / (1.0f + __expf(-x)); }

// ---------------------------------------------------------------------------
// WMMA (fp32, 16x16x4) GEMM:  C = A @ W^T + b1 + b2 (+ Cin)
// A: [M,K] row-major, W: [N,K] row-major, C: [M,N]
// REQUIRES: M % 16 == 0, N % 32 == 0, K % 4 == 0 (true for every call here).
// Each wave computes a 16x32 output strip (two 16x16 tiles sharing the A frag).
// No per-lane bounds checks -> no EXEC save/restore in the inner loop.
// ---------------------------------------------------------------------------
__global__ void gemm_wmma_kernel(const float* __restrict__ A, const float* __restrict__ W,
                                 const float* __restrict__ b1, const float* __restrict__ b2,
                                 const float* __restrict__ Cin, float* __restrict__ C,
                                 int M, int N, int K) {
    int wavesPerBlock = blockDim.x >> 5;
    int wave = blockIdx.x * wavesPerBlock + (threadIdx.x >> 5);
    int mtiles = M >> 4;
    int npairs = N >> 5;                 // strips of 32 columns
    if (wave >= mtiles * npairs) return; // wave-uniform branch
    int mt = wave / npairs;
    int np = wave - mt * npairs;
    int m0 = mt << 4, n0 = np << 5;

    int lane = threadIdx.x & 31;
    int half = lane >> 4;       // selects K pair (A/B) and M half (C/D)
    int r    = lane & 15;
    int na   = n0 + r;          // column for tile 0
    int nb   = na + 16;         // column for tile 1

    float biasA = 0.0f, biasB = 0.0f;
    if (b1) { biasA += b1[na]; biasB += b1[nb]; }
    if (b2) { biasA += b2[na]; biasB += b2[nb]; }

    v8f acc0, acc1;
#pragma unroll
    for (int q = 0; q < 8; ++q) {
        int m = m0 + q + half * 8;
        float v0 = biasA, v1 = biasB;
        if (Cin) {
            v0 += Cin[(long)m * N + na];
            v1 += Cin[(long)m * N + nb];
        }
        acc0[q] = v0;
        acc1[q] = v1;
    }

    const float* Arow = A + (long)(m0 + r) * K;
    const float* Wrow0 = W + (long)na * K;
    const float* Wrow1 = W + (long)nb * K;

#pragma unroll 2
    for (int k = 0; k < K; k += 4) {
        int kk = k + half * 2;
        v2f a;  a.x  = Arow[kk];   a.y  = Arow[kk + 1];
        v2f w0; w0.x = Wrow0[kk];  w0.y = Wrow0[kk + 1];
        v2f w1; w1.x = Wrow1[kk];  w1.y = Wrow1[kk + 1];
        acc0 = __builtin_amdgcn_wmma_f32_16x16x4_f32(false, a, false, w0, (short)0, acc0,
                                                     false, false);
        acc1 = __builtin_amdgcn_wmma_f32_16x16x4_f32(false, a, false, w1, (short)0, acc1,
                                                     false, false);
    }

#pragma unroll
    for (int q = 0; q < 8; ++q) {
        int m = m0 + q + half * 8;
        C[(long)m * N + na] = acc0[q];
        C[(long)m * N + nb] = acc1[q];
    }
}

// ---------------------------------------------------------------------------
// Encoder embedding gather: X0[m,k] = emb_enc[text[m], k]   (m = s*B+b)
// ---------------------------------------------------------------------------
__global__ void embed_enc_kernel(const int* __restrict__ text, const float* __restrict__ emb,
                                 float* __restrict__ X0) {
    int idx = blockIdx.x * blockDim.x + threadIdx.x;
    if (idx >= SLEN * BATCH * EMB) return;
    int m = idx >> 8;           // EMB == 256
    int k = idx & 255;
    X0[idx] = emb[(long)text[m] * EMB + k];
}

// ---------------------------------------------------------------------------
// Persistent single-workgroup LSTM recurrence (one layer, 512 steps).
// preG already holds x@Wih^T + bih + bhh for every timestep.
// LDS: h[16x512] (padded, rows 8..15 zero) | c[8x512] | gates[8x2048]
// 32 waves; each wave owns 4 N-tiles of the 8(->16) x 2048 gate GEMM.
// ---------------------------------------------------------------------------
__global__ void __launch_bounds__(1024)
lstm_seq_kernel(const float* __restrict__ preG, const float* __restrict__ Whh,
                float* __restrict__ hs_out, float* __restrict__ hfin,
                float* __restrict__ cfin, int S) {
    extern __shared__ float lds[];
    float* lh = lds;                       // 16*512
    float* lc = lds + 16 * HID;            // 8*512
    float* lg = lds + 16 * HID + 8 * HID;  // 8*2048

    int tid = threadIdx.x;
    for (int i = tid; i < 16 * HID; i += 1024) lh[i] = 0.0f;
    for (int i = tid; i < 8 * HID; i += 1024)  lc[i] = 0.0f;
    __syncthreads();

    int wave = tid >> 5, lane = tid & 31;
    int half = lane >> 4, r = lane & 15;

    for (int t = 0; t < S; ++t) {
        const float* pg = preG + (long)t * BATCH * G4H;
        // each wave: 2 strips of 2 N-tiles (4 tiles total), A frag shared per strip
        for (int i = 0; i < 2; ++i) {
            int n0 = (wave * 4 + i * 2) << 4;
            int na = n0 + r;
            int nb = na + 16;
            v8f acc0, acc1;
#pragma unroll
            for (int q = 0; q < 8; ++q) {
                int m = q + half * 8;
                bool valid = (m < BATCH);
                acc0[q] = valid ? pg[m * G4H + na] : 0.0f;
                acc1[q] = valid ? pg[m * G4H + nb] : 0.0f;
            }
            const float* hrow = lh + r * HID;
            const float* w0 = Whh + (long)na * HID;
            const float* w1 = Whh + (long)nb * HID;
#pragma unroll 2
            for (int k = 0; k < HID; k += 4) {
                int kk = k + half * 2;
                v2f a;  a.x  = hrow[kk]; a.y  = hrow[kk + 1];
                v2f b0; b0.x = w0[kk];   b0.y = w0[kk + 1];
                v2f b1v; b1v.x = w1[kk]; b1v.y = w1[kk + 1];
                acc0 = __builtin_amdgcn_wmma_f32_16x16x4_f32(false, a, false, b0, (short)0,
                                                             acc0, false, false);
                acc1 = __builtin_amdgcn_wmma_f32_16x16x4_f32(false, a, false, b1v, (short)0,
                                                             acc1, false, false);
            }
            if (half == 0) {               // rows 0..7 are the valid batch rows
#pragma unroll
                for (int q = 0; q < 8; ++q) {
                    lg[q * G4H + na] = acc0[q];
                    lg[q * G4H + nb] = acc1[q];
                }
            }
        }
        __syncthreads();

        for (int idx = tid; idx < BATCH * HID; idx += 1024) {
            int row = idx >> 9, j = idx & 511;
            float ig = lg[row * G4H + j];
            float fg = lg[row * G4H + HID + j];
            float gg = lg[row * G4H + 2 * HID + j];
            float og = lg[row * G4H + 3 * HID + j];
            float c2 = sigm(fg) * lc[idx] + sigm(ig) * tanhf(gg);
            float h2 = sigm(og) * tanhf(c2);
            lc[idx] = c2;
            lh[row * HID + j] = h2;
            if (hs_out) hs_out[(long)t * BATCH * HID + idx] = h2;
        }
        __syncthreads();
    }
    for (int idx = tid; idx < BATCH * HID; idx += 1024) {
        hfin[idx] = lh[idx];
        cfin[idx] = lc[idx];
    }
}

// ---------------------------------------------------------------------------
// Decoder helpers
// ---------------------------------------------------------------------------
__global__ void init_beam_kernel(const float* __restrict__ h1, const float* __restrict__ c1,
                                 float* __restrict__ dh, float* __restrict__ dc,
                                 int* __restrict__ curTok) {
    int idx = blockIdx.x * blockDim.x + threadIdx.x;
    if (idx < NBEAM * BATCH) curTok[idx] = 1;   // SOS
    if (idx >= NBEAM * BATCH * HID) return;
    int m = idx >> 9, j = idx & 511;
    int b = m & 7;
    dh[idx] = h1[b * HID + j];
    dc[idx] = c1[b * HID + j];
}

__global__ void gather_dec_embed_kernel(const int* __restrict__ curTok,
                                        const float* __restrict__ emb,
                                        float* __restrict__ x) {
    int idx = blockIdx.x * blockDim.x + threadIdx.x;
    if (idx >= NBEAM * BATCH * EMB) return;
    int m = idx >> 8, k = idx & 255;
    x[idx] = emb[(long)curTok[m] * EMB + k];
}

__global__ void dec_cell_kernel(const float* __restrict__ G, const float* __restrict__ c_in,
                                float* __restrict__ nh, float* __restrict__ nc) {
    int idx = blockIdx.x * blockDim.x + threadIdx.x;
    if (idx >= NBEAM * BATCH * HID) return;
    int m = idx >> 9, j = idx & 511;
    float ig = G[m * G4H + j];
    float fg = G[m * G4H + HID + j];
    float gg = G[m * G4H + 2 * HID + j];
    float og = G[m * G4H + 3 * HID + j];
    float c2 = sigm(fg) * c_in[idx] + sigm(ig) * tanhf(gg);
    nh[idx] = sigm(og) * tanhf(c2);
    nc[idx] = c2;
}

// single-head attention over L+1 = 3 keys; one block per (beam,b) row
__global__ void attn_kernel(const float* __restrict__ q, const float* __restrict__ kd,
                            const float* __restrict__ vd, const float* __restrict__ kenc,
                            const float* __restrict__ venc, float* __restrict__ ctx) {
    int m = blockIdx.x;           // m = beam*8 + b
    int b = m & 7;
    __shared__ float red[3][256];
    __shared__ float w[3];
    float s0 = 0.f, s1 = 0.f, s2 = 0.f;
    for (int j = threadIdx.x; j < HID; j += blockDim.x) {
        float qv = q[m * HID + j];
        s0 += qv * kenc[b * HID + j];
        s1 += qv * kenc[(BATCH + b) * HID + j];
        s2 += qv * kd[m * HID + j];
    }
    red[0][threadIdx.x] = s0; red[1][threadIdx.x] = s1; red[2][threadIdx.x] = s2;
    __syncthreads();
    for (int st = 128; st > 0; st >>= 1) {
        if (threadIdx.x < st) {
            red[0][threadIdx.x] += red[0][threadIdx.x + st];
            red[1][threadIdx.x] += red[1][threadIdx.x + st];
            red[2][threadIdx.x] += red[2][threadIdx.x + st];
        }
        __syncthreads();
    }
    if (threadIdx.x == 0) {
        const float sc = 0.04419417382415922f;   // 1/sqrt(512)
        float a0 = red[0][0] * sc, a1 = red[1][0] * sc, a2 = red[2][0] * sc;
        float mx = fmaxf(a0, fmaxf(a1, a2));
        float e0 = __expf(a0 - mx), e1 = __expf(a1 - mx), e2 = __expf(a2 - mx);
        float inv = 1.0f / (e0 + e1 + e2);
        w[0] = e0 * inv; w[1] = e1 * inv; w[2] = e2 * inv;
    }
    __syncthreads();
    for (int j = threadIdx.x; j < HID; j += blockDim.x) {
        ctx[m * HID + j] = w[0] * venc[b * HID + j] +
                           w[1] * venc[(BATCH + b) * HID + j] +
                           w[2] * vd[m * HID + j];
    }
}

// per-(beam,b) top-4 over 32000 logits; ties -> lower vocab index
__global__ void topk_kernel(const float* __restrict__ logits, float* __restrict__ topv,
                            int* __restrict__ topi) {
    int m = blockIdx.x;
    const float* row = logits + (long)m * VOC;
    float lv[4] = {-INFINITY, -INFINITY, -INFINITY, -INFINITY};
    int   li[4] = {INT_MAX, INT_MAX, INT_MAX, INT_MAX};
    for (int i = threadIdx.x; i < VOC; i += 256) {
        float v = row[i];
        for (int k = 0; k < 4; ++k) {
            if (v > lv[k] || (v == lv[k] && i < li[k])) {
                for (int j = 3; j > k; --j) { lv[j] = lv[j - 1]; li[j] = li[j - 1]; }
                lv[k] = v; li[k] = i;
                break;
            }
        }
    }
    __shared__ float sv[1024];
    __shared__ int   si[1024];
    for (int k = 0; k < 4; ++k) { sv[threadIdx.x * 4 + k] = lv[k]; si[threadIdx.x * 4 + k] = li[k]; }
    __syncthreads();
    if (threadIdx.x == 0) {
        for (int sel = 0; sel < 4; ++sel) {
            float bestv = -INFINITY; int besti = INT_MAX; int bestp = 0;
            for (int i = 0; i < 1024; ++i) {
                if (sv[i] > bestv || (sv[i] == bestv && si[i] < besti)) {
                    bestv = sv[i]; besti = si[i]; bestp = i;
                }
            }
            topv[m * 4 + sel] = bestv;
            topi[m * 4 + sel] = besti;
            sv[bestp] = -INFINITY;
        }
    }
}

// combine 16 beam-major candidates per batch row, record backpointers,
// gather new beam state (double-buffered).
__global__ void combine_kernel(const float* __restrict__ topv, const int* __restrict__ topi,
                               const float* __restrict__ nh, const float* __restrict__ nc,
                               float* __restrict__ dh_next, float* __restrict__ dc_next,
                               int* __restrict__ backptr_t, int* __restrict__ newtok_t,
                               int* __restrict__ curTok) {
    __shared__ int oldb_s[NBEAM * BATCH];
    int tid = threadIdx.x;
    if (tid < BATCH) {
        int b = tid;
        float cv[16]; int ct[16]; bool used[16];
        for (int n = 0; n < 4; ++n)
            for (int k = 0; k < 4; ++k) {
                cv[n * 4 + k] = topv[(n * BATCH + b) * 4 + k];
                ct[n * 4 + k] = topi[(n * BATCH + b) * 4 + k];
                used[n * 4 + k] = false;
            }
        for (int j = 0; j < 4; ++j) {
            int best = -1;
            for (int i = 0; i < 16; ++i)
                if (!used[i] && (best < 0 || cv[i] > cv[best])) best = i;
            used[best] = true;
            int oldb = best >> 2;
            int tok  = ct[best];
            int mrow = j * BATCH + b;
            backptr_t[mrow] = oldb;
            newtok_t[mrow]  = tok;
            curTok[mrow]    = tok;
            oldb_s[mrow]    = oldb;
        }
    }
    __syncthreads();
    for (int idx = tid; idx < NBEAM * BATCH * HID; idx += blockDim.x) {
        int m = idx >> 9, j = idx & 511;
        int src = oldb_s[m] * BATCH + (m & 7);
        dh_next[idx] = nh[src * HID + j];
        dc_next[idx] = nc[src * HID + j];
    }
}

// reconstruct beam-0 lineage; emit result tokens (as float)
__global__ void lineage_kernel(const int* __restrict__ backptr, const int* __restrict__ newtok,
                               float* __restrict__ out_res, int* __restrict__ linLog) {
    int b = threadIdx.x;
    if (b >= BATCH) return;
    int bm = 0;
    for (int t = NSTEP - 1; t >= 0; --t) {
        int m = bm * BATCH + b;
        out_res[b * TLEN + t + 1] = (float)newtok[t * NBEAM * BATCH + m];
        int parent = backptr[t * NBEAM * BATCH + m];
        linLog[t * BATCH + b] = parent;
        bm = parent;
    }
    out_res[b * TLEN + 0] = 1.0f;   // SOS
}

// output logits [B, V, T]; t=0 is onehot(1); t>=1 from stored raw logits via lineage
__global__ void write_logits_kernel(const float* __restrict__ lstore,
                                    const int* __restrict__ linLog,
                                    float* __restrict__ out) {
    long idx = (long)blockIdx.x * blockDim.x + threadIdx.x;
    if (idx >= (long)BATCH * VOC) return;
    int b = (int)(idx / VOC);
    int v = (int)(idx % VOC);
    float* dst = out + BATCH * TLEN + ((long)b * VOC + v) * TLEN;
    dst[0] = (v == 1) ? 1.0f : 0.0f;
    for (int t = 0; t < NSTEP; ++t) {
        int pb = linLog[t * BATCH + b];
        dst[t + 1] = lstore[((long)t * NBEAM * BATCH + pb * BATCH + b) * VOC + v];
    }
}

// ---------------------------------------------------------------------------
extern "C" void kernel_launch(void* const* d_in, const int* in_sizes, int n_in,
                              void* d_out, int out_size, void* d_ws, size_t ws_size,
                              hipStream_t stream) {
    (void)in_sizes; (void)n_in; (void)out_size; (void)ws_size;

    const int*   text     = (const int*)d_in[0];
    const float* emb_enc  = (const float*)d_in[1];
    const float* emb_dec  = (const float*)d_in[2];
    const float* eWih0    = (const float*)d_in[3];
    const float* eWhh0    = (const float*)d_in[4];
    const float* ebih0    = (const float*)d_in[5];
    const float* ebhh0    = (const float*)d_in[6];
    const float* eWih1    = (const float*)d_in[7];
    const float* eWhh1    = (const float*)d_in[8];
    const float* ebih1    = (const float*)d_in[9];
    const float* ebhh1    = (const float*)d_in[10];
    const float* dWih     = (const float*)d_in[11];
    const float* dWhh     = (const float*)d_in[12];
    const float* dbih     = (const float*)d_in[13];
    const float* dbhh     = (const float*)d_in[14];
    const float* Wq = (const float*)d_in[15]; const float* bq = (const float*)d_in[16];
    const float* Wk = (const float*)d_in[17]; const float* bk = (const float*)d_in[18];
    const float* Wv = (const float*)d_in[19]; const float* bv = (const float*)d_in[20];
    const float* Wo = (const float*)d_in[21]; const float* bo = (const float*)d_in[22];
    const float* Wout = (const float*)d_in[23]; const float* bout = (const float*)d_in[24];

    float* ws = (float*)d_ws;
    size_t o = 0;
    auto alloc = [&](size_t n) { float* p = ws + o; o += n; return p; };
    float* X0    = alloc((size_t)SLEN * BATCH * EMB);
    float* preG  = alloc((size_t)SLEN * BATCH * G4H);
    float* hs0   = alloc((size_t)SLEN * BATCH * HID);
    float* hfin  = alloc(2 * BATCH * HID);
    float* cfin  = alloc(2 * BATCH * HID);
    float* kenc  = alloc(2 * BATCH * HID);
    float* venc  = alloc(2 * BATCH * HID);
    float* dh[2] = { alloc(NBEAM * BATCH * HID), alloc(NBEAM * BATCH * HID) };
    float* dc[2] = { alloc(NBEAM * BATCH * HID), alloc(NBEAM * BATCH * HID) };
    float* nhb   = alloc(NBEAM * BATCH * HID);
    float* ncb   = alloc(NBEAM * BATCH * HID);
    float* Gdec  = alloc(NBEAM * BATCH * G4H);
    float* xdec  = alloc(NBEAM * BATCH * EMB);
    float* qb    = alloc(NBEAM * BATCH * HID);
    float* kdb   = alloc(NBEAM * BATCH * HID);
    float* vdb   = alloc(NBEAM * BATCH * HID);
    float* ctxb  = alloc(NBEAM * BATCH * HID);
    float* oproj = alloc(NBEAM * BATCH * HID);
    float* lstore = alloc((size_t)NSTEP * NBEAM * BATCH * VOC);
    float* topv  = alloc(NBEAM * BATCH * 4);
    int* topi    = (int*)(ws + o); o += NBEAM * BATCH * 4;
    int* backptr = (int*)(ws + o); o += NSTEP * NBEAM * BATCH;
    int* newtok  = (int*)(ws + o); o += NSTEP * NBEAM * BATCH;
    int* curTok  = (int*)(ws + o); o += NBEAM * BATCH;
    int* linLog  = (int*)(ws + o); o += NSTEP * BATCH;

    const int ldsBytes = (16 * HID + 8 * HID + 8 * G4H) * 4;   // 114688
    hipFuncSetAttribute((const void*)lstm_seq_kernel,
                        hipFuncAttributeMaxDynamicSharedMemorySize, ldsBytes);

    auto gemm = [&](const float* A, const float* W, const float* b1, const float* b2,
                    const float* Cin, float* C, int M, int N, int K) {
        int tiles = (M / 16) * (N / 32);     // 16x32 strips per wave
        int blocks = (tiles + 7) / 8;
        gemm_wmma_kernel<<<blocks, 256, 0, stream>>>(A, W, b1, b2, Cin, C, M, N, K);
    };

    // ---- encoder ----
    embed_enc_kernel<<<(SLEN * BATCH * EMB) / 256, 256, 0, stream>>>(text, emb_enc, X0);
    gemm(X0, eWih0, ebih0, ebhh0, nullptr, preG, SLEN * BATCH, G4H, EMB);
    lstm_seq_kernel<<<1, 1024, ldsBytes, stream>>>(preG, eWhh0, hs0, hfin, cfin, SLEN);
    gemm(hs0, eWih1, ebih1, ebhh1, nullptr, preG, SLEN * BATCH, G4H, HID);
    lstm_seq_kernel<<<1, 1024, ldsBytes, stream>>>(preG, eWhh1, nullptr,
                                                   hfin + BATCH * HID, cfin + BATCH * HID, SLEN);
    // enc_hidden = [h0 ; h1] is hfin [16, 512]
    gemm(hfin, Wk, bk, nullptr, nullptr, kenc, 2 * BATCH, HID, HID);
    gemm(hfin, Wv, bv, nullptr, nullptr, venc, 2 * BATCH, HID, HID);
    init_beam_kernel<<<64, 256, 0, stream>>>(hfin + BATCH * HID, cfin + BATCH * HID,
                                             dh[0], dc[0], curTok);

    // ---- beam-search decoder ----
    for (int t = 0; t < NSTEP; ++t) {
        int cur = t & 1, nxt = cur ^ 1;
        gather_dec_embed_kernel<<<32, 256, 0, stream>>>(curTok, emb_dec, xdec);
        gemm(xdec, dWih, dbih, dbhh, nullptr, Gdec, NBEAM * BATCH, G4H, EMB);
        gemm(dh[cur], dWhh, nullptr, nullptr, Gdec, Gdec, NBEAM * BATCH, G4H, HID);
        dec_cell_kernel<<<64, 256, 0, stream>>>(Gdec, dc[cur], nhb, ncb);
        gemm(nhb, Wq, bq, nullptr, nullptr, qb, NBEAM * BATCH, HID, HID);
        gemm(nhb, Wk, bk, nullptr, nullptr, kdb, NBEAM * BATCH, HID, HID);
        gemm(nhb, Wv, bv, nullptr, nullptr, vdb, NBEAM * BATCH, HID, HID);
        attn_kernel<<<NBEAM * BATCH, 256, 0, stream>>>(qb, kdb, vdb, kenc, venc, ctxb);
        gemm(ctxb, Wo, bo, nullptr, nullptr, oproj, NBEAM * BATCH, HID, HID);
        float* lt = lstore + (size_t)t * NBEAM * BATCH * VOC;
        gemm(oproj, Wout, bout, nullptr, nullptr, lt, NBEAM * BATCH, VOC, HID);
        topk_kernel<<<NBEAM * BATCH, 256, 0, stream>>>(lt, topv, topi);
        combine_kernel<<<1, 256, 0, stream>>>(topv, topi, nhb, ncb, dh[nxt], dc[nxt],
                                              backptr + t * NBEAM * BATCH,
                                              newtok + t * NBEAM * BATCH, curTok);
    }

    // ---- finalize outputs ----
    lineage_kernel<<<1, 32, 0, stream>>>(backptr, newtok, (float*)d_out, linLog);
    write_logits_kernel<<<(BATCH * VOC + 255) / 256, 256, 0, stream>>>(lstore, linLog,
                                                                       (float*)d_out);
}